// Model_3599182594439
// MI455X (gfx1250) — compile-verified
//
#include <hip/hip_runtime.h>

typedef __attribute__((ext_vector_type(2))) float v2f;
typedef __attribute__((ext_vector_type(4))) float v4f;
typedef __attribute__((ext_vector_type(8))) float v8f;

#define IMG     256
#define NFACES  512
#define INV_SIGMA 100.0f
#define EPSF    1e-8f
#define LOG1EM6 -13.815511f   // logf(1e-6)
#define LOG2E   1.442695041f
#define LN2     0.6931471806f

// ---------------------------------------------------------------------------
// Kernel 1: camera transform + per-face edge line coefficients.
// Edge k of face f stored in Ek[f] as float4 (cA, cB, cC, 0) with
//   d(px,py)/SIGMA = cA*px + cB*py + cC     (inv_len * sgn * 1/SIGMA folded in;
//                                            legal since min() commutes with
//                                            positive scaling)
// float4 layout matches the WMMA B-fragment split: lanes 0-15 load .xy
// (B rows K=0,1), lanes 16-31 load .zw (B rows K=2,3=0).
// ---------------------------------------------------------------------------
__global__ void edge_setup(const float* __restrict__ verts,
                           const int*   __restrict__ faces,
                           float* __restrict__ E0,
                           float* __restrict__ E1,
                           float* __restrict__ E2)
{
    int f = blockIdx.x * blockDim.x + threadIdx.x;
    if (f >= NFACES) return;

    const float D2R = 0.017453292519943295f;
    // eye = get_eye(2.732, el=0, az=90deg)
    float az = 90.0f * D2R;
    float eyex = 2.732f * __sinf(az);
    float eyey = 0.0f;
    float eyez = -2.732f * __cosf(az);
    // z = normalize(at - eye) = normalize(-eye)
    float zx = -eyex, zy = -eyey, zz = -eyez;
    float zn = sqrtf(zx*zx + zy*zy + zz*zz) + EPSF;
    zx /= zn; zy /= zn; zz /= zn;
    // x = normalize(cross(up, z)), up = (0,1,0) -> (zz, 0, -zx)
    float xx = zz, xy = 0.0f, xz = -zx;
    float xn = sqrtf(xx*xx + xy*xy + xz*xz) + EPSF;
    xx /= xn; xy /= xn; xz /= xn;
    // y = normalize(cross(z, x))
    float yx = zy*xz - zz*xy;
    float yy = zz*xx - zx*xz;
    float yz = zx*xy - zy*xx;
    float yn = sqrtf(yx*yx + yy*yy + yz*yz) + EPSF;
    yx /= yn; yy /= yn; yz /= yn;

    float t = tanf(30.0f * D2R);

    float sx[3], sy[3];
    #pragma unroll
    for (int k = 0; k < 3; ++k) {
        int vi = faces[f * 3 + k];
        float wx = verts[vi * 3 + 0] - eyex;
        float wy = verts[vi * 3 + 1] - eyey;
        float wz = verts[vi * 3 + 2] - eyez;
        float cx = wx*xx + wy*xy + wz*xz;
        float cy = wx*yx + wy*yy + wz*yz;
        float cz = wx*zx + wy*zy + wz*zz;
        float den = cz * t + EPSF;
        sx[k] = cx / den;
        sy[k] = cy / den;
    }
    float area = (sx[1]-sx[0])*(sy[2]-sy[0]) - (sy[1]-sy[0])*(sx[2]-sx[0]);
    float sgn = (area >= 0.0f) ? 1.0f : -1.0f;

    float* E[3] = {E0, E1, E2};
    const int nxt[3] = {1, 2, 0};
    #pragma unroll
    for (int k = 0; k < 3; ++k) {
        float p0x = sx[k],      p0y = sy[k];
        float p1x = sx[nxt[k]], p1y = sy[nxt[k]];
        float ex = p1x - p0x, ey = p1y - p0y;
        float s = rsqrtf(ex*ex + ey*ey + EPSF) * sgn * INV_SIGMA;
        v4f out;
        out.x = -ey * s;                  // coeff of px
        out.y =  ex * s;                  // coeff of py
        out.z = (ey*p0x - ex*p0y) * s;    // constant
        out.w = 0.0f;
        *(v4f*)(E[k] + f * 4) = out;
    }
}

// ---------------------------------------------------------------------------
// Kernel 2: 1 wave = 16 pixels. 32 chunks of 16 faces; per chunk 3x
// V_WMMA_F32_16X16X4_F32 (one per edge-slot) -> v_min3 -> softplus via raw
// v_exp_f32 / v_log_f32 (argument of log is in (1,2], no denorm guard needed).
// Face-dim reduced with shfl_xor, block partial via LDS.
// ---------------------------------------------------------------------------
__global__ void sil_loss(const float* __restrict__ E0,
                         const float* __restrict__ E1,
                         const float* __restrict__ E2,
                         const float* __restrict__ ref,
                         float* __restrict__ partials)
{
    __shared__ float sacc[8];
    const int lane = threadIdx.x & 31;
    const int wave = threadIdx.x >> 5;
    const int group = blockIdx.x * 8 + wave;      // 16-pixel group id, 0..4095
    const int m   = lane & 15;
    const int hi  = lane >> 4;                    // 0: low half, 1: high half
    const int pix0 = group * 16;

    // A fragment: rows = 16 pixels, K = [px, py, 1, 0]
    const int p   = pix0 + m;
    const int row = p >> 8, col = p & 255;
    const float px = (col + 0.5f) * (2.0f / IMG) - 1.0f;
    const float py = -((row + 0.5f) * (2.0f / IMG) - 1.0f);
    v2f afrag;
    afrag.x = (lane < 16) ? px : 1.0f;    // lanes 0-15: K=0,1 ; lanes 16-31: K=2,3
    afrag.y = (lane < 16) ? py : 0.0f;

    const int boff = m * 4 + hi * 2;      // float offset into Ek (float4 per face)

    v8f acc = {};
    v8f zero = {};
    #pragma unroll 2
    for (int ch = 0; ch < 32; ++ch) {
        const int fb4 = ch * 16 * 4;
        v2f b0 = *(const v2f*)(E0 + fb4 + boff);
        v2f b1 = *(const v2f*)(E1 + fb4 + boff);
        v2f b2 = *(const v2f*)(E2 + fb4 + boff);
        v8f d0 = __builtin_amdgcn_wmma_f32_16x16x4_f32(false, afrag, false, b0,
                                                       (short)0, zero, false, false);
        v8f d1 = __builtin_amdgcn_wmma_f32_16x16x4_f32(false, afrag, false, b1,
                                                       (short)0, zero, false, false);
        v8f d2 = __builtin_amdgcn_wmma_f32_16x16x4_f32(false, afrag, false, b2,
                                                       (short)0, zero, false, false);
        #pragma unroll
        for (int i = 0; i < 8; ++i) {
            // x = dmin/SIGMA (scale folded into edge coeffs)
            float x  = fminf(fminf(d0[i], d1[i]), d2[i]);
            // log(1 - sigmoid(x)) = -softplus(x)
            //                     = -max(x,0) - ln2 * log2(1 + exp2(-log2e*|x|))
            float t  = __builtin_amdgcn_exp2f(-LOG2E * fabsf(x));
            float lg = __builtin_amdgcn_logf(1.0f + t);   // v_log_f32: arg in (1,2]
            float mx = fmaxf(x, 0.0f);
            float l  = fmaf(-LN2, lg, -mx);
            acc[i] += fmaxf(l, LOG1EM6);                  // clip(alpha <= 1-1e-6)
        }
    }

    // Reduce face dimension: D layout puts face N in lane (N + 16*hi);
    // sum across the 16 lanes of each half, then finish per pixel.
    float waccum = 0.0f;
    #pragma unroll
    for (int i = 0; i < 8; ++i) {
        float v = acc[i];
        v += __shfl_xor(v, 1, 32);
        v += __shfl_xor(v, 2, 32);
        v += __shfl_xor(v, 4, 32);
        v += __shfl_xor(v, 8, 32);
        // pixel M = i (low half lanes) or i+8 (high half lanes)
        int pp = pix0 + i + hi * 8;
        float sil  = 1.0f - __expf(v);
        float diff = sil - ref[pp];
        waccum += diff * diff;
    }
    // unique values live in lanes 0 and 16
    waccum = (m == 0) ? waccum : 0.0f;
    waccum += __shfl_xor(waccum, 16, 32);
    if (lane == 0) sacc[wave] = waccum;
    __syncthreads();
    if (threadIdx.x == 0) {
        float s = 0.0f;
        #pragma unroll
        for (int w = 0; w < 8; ++w) s += sacc[w];
        partials[blockIdx.x] = s;
    }
}

// ---------------------------------------------------------------------------
// Kernel 3: deterministic tree reduction of 512 block partials -> loss scalar.
// ---------------------------------------------------------------------------
__global__ void final_reduce(const float* __restrict__ partials,
                             float* __restrict__ out)
{
    __shared__ float s[256];
    int t = threadIdx.x;
    s[t] = partials[t] + partials[t + 256];
    __syncthreads();
    for (int stride = 128; stride > 0; stride >>= 1) {
        if (t < stride) s[t] += s[t + stride];
        __syncthreads();
    }
    if (t == 0) out[0] = s[0];
}

extern "C" void kernel_launch(void* const* d_in, const int* in_sizes, int n_in,
                              void* d_out, int out_size, void* d_ws, size_t ws_size,
                              hipStream_t stream) {
    const float* verts = (const float*)d_in[0];   // (1, 4096, 3) f32
    const int*   faces = (const int*)d_in[1];     // (1, 512, 3)  i32
    const float* ref   = (const float*)d_in[2];   // (256, 256)   f32

    float* E0       = (float*)d_ws;               // 512 * float4
    float* E1       = E0 + NFACES * 4;
    float* E2       = E1 + NFACES * 4;
    float* partials = E2 + NFACES * 4;            // 512 floats

    edge_setup<<<2, 256, 0, stream>>>(verts, faces, E0, E1, E2);
    sil_loss<<<512, 256, 0, stream>>>(E0, E1, E2, ref, partials);
    final_reduce<<<1, 256, 0, stream>>>(partials, (float*)d_out);
}